// LSTM_266287972489
// MI455X (gfx1250) — compile-verified
//
#include <hip/hip_runtime.h>
#include <hip/hip_bf16.h>
#include <math.h>

typedef __attribute__((ext_vector_type(2))) float v2f;
typedef __attribute__((ext_vector_type(8))) float v8f;

#define HID 100
#define G4  400   // 4 * HID gate rows

// ---------------------------------------------------------------------------
// Kernel A: x_gates[t][j] = sum_k input[t][k] * W_ih[j][k] + (b_ih[j]+b_hh[j])
// One V_WMMA_F32_16X16X4_F32 per 16x16 tile (K=4 exactly).
// A layout (f32 16x4): lane L<16 holds row M=L, VGPR0=K0,VGPR1=K1;
//                      lane L+16 holds row M=L, VGPR0=K2,VGPR1=K3.
// B layout (f32 4x16): lane L<16 holds col N=L, VGPR0=K0,VGPR1=K1;
//                      lane L+16 holds col N=L, VGPR0=K2,VGPR1=K3.
// C/D layout: VGPR r, lane L -> row (r + 8*(L>>4)), col (L&15).
// ---------------------------------------------------------------------------
__global__ __launch_bounds__(256)
void xgate_wmma_kernel(const float* __restrict__ input,  // [S,4]
                       const float* __restrict__ Wih,    // [400,4]
                       const float* __restrict__ bih,    // [400]
                       const float* __restrict__ bhh,    // [400]
                       float* __restrict__ xg,           // [S,400]
                       int S)
{
  const int lane = threadIdx.x & 31;
  const int wave = blockIdx.x * (blockDim.x >> 5) + (threadIdx.x >> 5);
  const int t0 = wave * 16;
  if (t0 >= S) return;

  const int l15  = lane & 15;
  const int half = lane >> 4;

  // A tile: 16 timesteps x 4 inputs
  float2 af = *(const float2*)(input + (size_t)(t0 + l15) * 4 + 2 * half);
  v2f a; a[0] = af.x; a[1] = af.y;

  for (int jt = 0; jt < G4 / 16; ++jt) {
    const int j0 = jt * 16;
    const int j  = j0 + l15;

    // B tile: B[k][n] = W_ih[j0+n][k]; W_ih row-major [400][4]
    float2 bf = *(const float2*)(Wih + (size_t)j * 4 + 2 * half);
    v2f b; b[0] = bf.x; b[1] = bf.y;

    const float bsum = bih[j] + bhh[j];   // bias depends only on column
    v8f c;
#pragma unroll
    for (int r = 0; r < 8; ++r) c[r] = bsum;

    v8f d = __builtin_amdgcn_wmma_f32_16x16x4_f32(
        /*neg_a=*/false, a, /*neg_b=*/false, b,
        /*c_mod=*/(short)0, c, /*reuse_a=*/false, /*reuse_b=*/false);

#pragma unroll
    for (int r = 0; r < 8; ++r) {
      xg[(size_t)(t0 + r + 8 * half) * G4 + j] = d[r];
    }
  }
}

// ---------------------------------------------------------------------------
// Kernel B: the sequential recurrence. Single workgroup, 416 threads (13
// wave32 waves). Thread j < 400 owns gate row j with W_hh[j][0..99] resident
// in 100 VGPRs. h lives in LDS (broadcast float4 reads, conflict-free).
// Two barriers per step. c[j] lives in the register of thread j (<100).
// USE_XG=1: x_gates streamed from d_ws with a depth-4 register prefetch.
// USE_XG=0: x contribution computed inline (4 FMAs) from a prefetched float4.
// ---------------------------------------------------------------------------
template <int USE_XG>
__global__ __launch_bounds__(416, 1)
void lstm_seq_kernel(const float* __restrict__ input,  // [S,4]
                     const float* __restrict__ Wih,    // [400,4]
                     const float* __restrict__ Whh,    // [400,100]
                     const float* __restrict__ bih,    // [400]
                     const float* __restrict__ bhh,    // [400]
                     const float* __restrict__ Wlin,   // [1,100]
                     const float* __restrict__ blin,   // [1]
                     const float* __restrict__ xg,     // [S,400] or null
                     float* __restrict__ out,          // [1]
                     int S)
{
  __shared__ float h_s[HID];
  __shared__ float gate_s[G4];

  const int  tid    = threadIdx.x;
  const bool active = tid < G4;
  const int  j      = active ? tid : 0;

  // Pin W_hh row j into registers (fully unrolled -> no spills to LDS/indexing)
  float w[HID];
#pragma unroll
  for (int kk = 0; kk < HID / 4; ++kk) {
    float4 wv = *(const float4*)(Whh + (size_t)j * HID + 4 * kk);
    w[4 * kk + 0] = wv.x; w[4 * kk + 1] = wv.y;
    w[4 * kk + 2] = wv.z; w[4 * kk + 3] = wv.w;
  }
  float4 wx = *(const float4*)(Wih + (size_t)j * 4);
  const float bsum = bih[j] + bhh[j];
  const bool  is_g = (tid >= 200 && tid < 300);   // tanh gate rows

  float c = 0.0f;
  if (tid < HID) h_s[tid] = 0.0f;
  __syncthreads();

  // Depth-4 register prefetch pipeline for the per-step input contribution.
  const int D = 4;
  float4 xq[D];
  float  gq[D];
#pragma unroll
  for (int u = 0; u < D; ++u) {
    if (u < S) {
      if (USE_XG) gq[u] = xg[(size_t)u * G4 + j];
      else        xq[u] = *(const float4*)(input + (size_t)u * 4);
    }
  }

  for (int t0 = 0; t0 < S; t0 += D) {
#pragma unroll
    for (int u = 0; u < D; ++u) {
      const int t = t0 + u;
      if (t >= S) break;                 // uniform across block

      float acc;
      if (USE_XG) {
        acc = gq[u];
      } else {
        float4 x = xq[u];
        acc = bsum;
        acc = fmaf(wx.x, x.x, acc);
        acc = fmaf(wx.y, x.y, acc);
        acc = fmaf(wx.z, x.z, acc);
        acc = fmaf(wx.w, x.w, acc);
      }

      // Prefetch step t+D (consumed one chunk later; hides HBM/L2 latency)
      const int tp = t + D;
      if (tp < S) {
        if (USE_XG) gq[u] = xg[(size_t)tp * G4 + j];
        else        xq[u] = *(const float4*)(input + (size_t)tp * 4);
      }

      // gates[j] += h . W_hh[j,:]  (h broadcast from LDS, w in VGPRs)
#pragma unroll
      for (int kk = 0; kk < HID / 4; ++kk) {
        float4 hv = *(const float4*)(h_s + 4 * kk);
        acc = fmaf(w[4 * kk + 0], hv.x, acc);
        acc = fmaf(w[4 * kk + 1], hv.y, acc);
        acc = fmaf(w[4 * kk + 2], hv.z, acc);
        acc = fmaf(w[4 * kk + 3], hv.w, acc);
      }

      // Activation without divergence: tanh(x) = 2*sigmoid(2x) - 1
      float ax = is_g ? 2.0f * acc : acc;
      float sg = 1.0f / (1.0f + expf(-ax));
      float av = is_g ? fmaf(2.0f, sg, -1.0f) : sg;
      if (active) gate_s[tid] = av;
      __syncthreads();

      if (tid < HID) {
        float gi = gate_s[tid];
        float gf = gate_s[tid + 100];
        float gg = gate_s[tid + 200];
        float go = gate_s[tid + 300];
        c = fmaf(gf, c, gi * gg);
        // tanh(c) via expf (saturates correctly for |c| large)
        float e  = expf(-2.0f * c);
        float th = fmaf(2.0f, 1.0f / (1.0f + e), -1.0f);
        h_s[tid] = go * th;
      }
      __syncthreads();
    }
  }

  // Final linear: scalar output, done once.
  if (tid == 0) {
    float s = blin[0];
    for (int k = 0; k < HID; ++k) s = fmaf(h_s[k], Wlin[k], s);
    out[0] = s;
  }
}

// ---------------------------------------------------------------------------
extern "C" void kernel_launch(void* const* d_in, const int* in_sizes, int n_in,
                              void* d_out, int out_size, void* d_ws, size_t ws_size,
                              hipStream_t stream)
{
  const float* input = (const float*)d_in[0];
  const float* Wih   = (const float*)d_in[1];
  const float* Whh   = (const float*)d_in[2];
  const float* bih   = (const float*)d_in[3];
  const float* bhh   = (const float*)d_in[4];
  const float* Wlin  = (const float*)d_in[5];
  const float* blin  = (const float*)d_in[6];
  float* out = (float*)d_out;

  const int    S    = in_sizes[0] / 4;
  const size_t need = (size_t)S * G4 * sizeof(float);
  const bool   use_xg = (ws_size >= need) && (S % 16 == 0);

  if (use_xg) {
    float* xg = (float*)d_ws;
    const int rowTiles      = S / 16;
    const int wavesPerBlock = 8;
    const int blocks        = (rowTiles + wavesPerBlock - 1) / wavesPerBlock;
    xgate_wmma_kernel<<<blocks, 32 * wavesPerBlock, 0, stream>>>(
        input, Wih, bih, bhh, xg, S);
    lstm_seq_kernel<1><<<1, 416, 0, stream>>>(
        input, Wih, Whh, bih, bhh, Wlin, blin, xg, out, S);
  } else {
    lstm_seq_kernel<0><<<1, 416, 0, stream>>>(
        input, Wih, Whh, bih, bhh, Wlin, blin, nullptr, out, S);
  }
}